// IzhikevichNeuron_39032662786071
// MI455X (gfx1250) — compile-verified
//
#include <hip/hip_runtime.h>
#include <stdint.h>

// ---------------------------------------------------------------------------
// Izhikevich neuron scan, MI455X (gfx1250).
//
// Roofline: 256 MB read + 256 MB write => ~22 us @ 23.3 TB/s. ~12 VALU ops per
// element => comparable VALU time. Balanced streaming kernel; the CDNA5 path
// that matters is the Tensor Data Mover: each block's input slab is a 2D tile
// (CHUNK_T rows x TILE_N contiguous floats, row stride N), DMA'd into an
// 8-deep LDS ring. 256 blocks x 8 tiles x 8 KB = 16 MB in flight, enough to
// cover HBM latency at 23.3 TB/s with zero VGPR cost. Input reads and spike
// stores are both non-temporal: each byte is touched exactly once.
// ---------------------------------------------------------------------------

namespace {
constexpr int kT      = 2048;            // timesteps (reference-fixed)
constexpr int kN      = 32768;           // neurons   (reference-fixed)
constexpr int kTileN  = 128;             // neurons per block (== blockDim.x)
constexpr int kChunkT = 16;              // timesteps per TDM tile
constexpr int kDepth  = 8;               // pipeline depth (LDS ring slots)
constexpr int kChunks = kT / kChunkT;    // 128 chunks
constexpr int kSlotFloats = kChunkT * kTileN;                 // 2048 floats = 8 KB
constexpr unsigned kSlotBytes = kSlotFloats * sizeof(float);  // 8192
constexpr size_t kLdsBytes = (size_t)kDepth * kSlotBytes;     // 64 KB of 320 KB
}  // namespace

typedef uint32_t u32x4 __attribute__((ext_vector_type(4)));
typedef uint32_t u32x8 __attribute__((ext_vector_type(8)));

// Low 32 bits of a generic pointer into LDS == wave-relative LDS byte offset
// (ISA 10.2: LDS aperture address truncates to addr[31:0]).
__device__ __forceinline__ uint32_t lds_byte_offset(const void* p) {
  return (uint32_t)(uintptr_t)p;
}

// Issue one TDM 2D tile load: kChunkT rows x kTileN floats, row stride kN.
// D# layout per CDNA5 ISA ch.8 (group0: count/lds_addr/global_addr/type=2;
// group1: data_size, tensor dims, tile dims, strides; groups 2/3 zero => dims
// 2..4 unused). Issued by one wave; tracked with TENSORcnt. th:TH_LOAD_NT:
// stream-once input, keep it out of L2 residency.
__device__ __forceinline__ void tdm_load_tile(const float* gsrc, uint32_t lds_off) {
  const uint64_t ga = (uint64_t)(uintptr_t)gsrc;

  u32x4 g0;
  g0[0] = 1u;                                                // count=1, no gather
  g0[1] = lds_off;                                           // lds_addr (bytes)
  g0[2] = (uint32_t)ga;                                      // global_addr[31:0]
  g0[3] = (uint32_t)((ga >> 32) & 0x01FFFFFFull)             // global_addr[56:32]
          | 0x80000000u;                                     // type=2 ("image")

  constexpr uint64_t s0 = (uint64_t)kN;                      // dim0 stride (elems)
  constexpr uint64_t s1 = (uint64_t)kN * (uint64_t)kT;       // dim1 stride (unused: tile_dim2=0)
  u32x8 g1;
  g1[0] = 0x00020000u;                                       // data_size=2 (4B), no pad/iter/mask
  g1[1] = ((uint32_t)(kN & 0xFFFF)) << 16;                   // tensor_dim0[15:0]<<16 (bar addr=0)
  g1[2] = ((uint32_t)kN >> 16) | (((uint32_t)(kT & 0xFFFF)) << 16);  // dim0 hi | dim1 lo
  g1[3] = ((uint32_t)kT >> 16) | ((uint32_t)kTileN << 16);   // dim1 hi | tile_dim0
  g1[4] = (uint32_t)kChunkT;                                 // tile_dim1 (tile_dim2=0)
  g1[5] = (uint32_t)(s0 & 0xFFFFFFFFull);                    // tensor_dim0_stride[31:0]
  g1[6] = (uint32_t)((s0 >> 32) & 0xFFFF) | (uint32_t)((s1 & 0xFFFF) << 16);
  g1[7] = (uint32_t)(s1 >> 16);                              // tensor_dim1_stride[47:16]

  u32x4 gz = {0u, 0u, 0u, 0u};                               // groups 2/3: dims 2..4 unused

  asm volatile("tensor_load_to_lds %0, %1, %2, %3 th:TH_LOAD_NT"
               :
               : "s"(g0), "s"(g1), "s"(gz), "s"(gz)
               : "memory");
}

// One Izhikevich step; returns spike (0/1) as float.
__device__ __forceinline__ float izh_step(float& v, float& u, float It) {
  // dv = 0.04 v^2 + 5 v + 140 - u + I ; du = 0.02 (0.2 v - u)   (old v,u)
  float dv = __builtin_fmaf(0.04f * v, v,
             __builtin_fmaf(5.0f, v, (140.0f - u) + It));
  float du = 0.02f * __builtin_fmaf(0.2f, v, -u);
  v = __builtin_fmaf(0.2f, dv, v);   // v += DT*dv
  u = __builtin_fmaf(0.2f, du, u);   // u += DT*du
  const bool spk = (v >= 30.0f);
  const float s = spk ? 1.0f : 0.0f;
  v = spk ? 0.06f : v;               // reset to C
  u = spk ? (u + 6.0f) : u;          // u += D
  return s;
}

// Consume one chunk's tile from LDS: 16 neuron steps + spike stores.
__device__ __forceinline__ void consume_chunk(const float* smem, int c, int tid,
                                              float& v, float& u, float*& outp) {
  const float* slot = smem + (c & (kDepth - 1)) * kSlotFloats + tid;
#pragma unroll
  for (int tt = 0; tt < kChunkT; ++tt) {
    const float It = slot[tt * kTileN];        // ds_load_2addr_b32 pairs, conflict-free
    const float s  = izh_step(v, u, It);
    __builtin_nontemporal_store(s, outp);      // stream-once output, bypass L2 residency
    outp += kN;
  }
}

// Epilogue drain with compile-time wait immediates (s_wait_tensorcnt requires
// a constant integer argument, so unroll via template recursion).
template <int K>
__device__ __forceinline__ void drain_tail(bool issuer, const float* smem, int tid,
                                           float& v, float& u, float*& outp) {
  if constexpr (K < kDepth) {
    constexpr int c = kChunks - kDepth + K;
    if (issuer) __builtin_amdgcn_s_wait_tensorcnt(kDepth - 1 - K);
    __syncthreads();
    consume_chunk(smem, c, tid, v, u, outp);
    drain_tail<K + 1>(issuer, smem, tid, v, u, outp);
  }
}

__global__ __launch_bounds__(kTileN) void IzhikevichNeuron_39032662786071_kernel(
    const float* __restrict__ I, float* __restrict__ spikes) {
  extern __shared__ float smem[];  // kDepth ring of [kChunkT x kTileN] tiles

  const int tid = threadIdx.x;
  const int n0  = blockIdx.x * kTileN;   // first neuron of this block
  const int n   = n0 + tid;              // this thread's neuron
  const bool issuer = (tid < 32);        // wave 0 drives the TDM pipeline

  // Prologue: fill the ring with chunks 0..kDepth-1.
  if (issuer) {
    const uint32_t base = lds_byte_offset(smem);
#pragma unroll
    for (int k = 0; k < kDepth; ++k) {
      tdm_load_tile(I + (size_t)k * kChunkT * kN + n0, base + (uint32_t)k * kSlotBytes);
    }
  }

  float v = 0.06f;           // V_INIT
  float u = 0.2f * 0.06f;    // B * V_INIT
  float* outp = spikes + n;  // advances by kN per timestep (coalesced per wave)

  // Steady state: wait oldest DMA -> barrier -> compute 16 steps -> barrier ->
  // reissue the freed slot kDepth chunks ahead.
  for (int c = 0; c < kChunks - kDepth; ++c) {
    if (issuer) __builtin_amdgcn_s_wait_tensorcnt(kDepth - 1);
    __syncthreads();  // tile c visible to all waves

    consume_chunk(smem, c, tid, v, u, outp);

    __syncthreads();  // all waves done reading slot before DMA overwrites it
    if (issuer) {
      tdm_load_tile(I + (size_t)(c + kDepth) * kChunkT * kN + n0,
                    lds_byte_offset(smem) + (uint32_t)(c & (kDepth - 1)) * kSlotBytes);
    }
  }

  // Drain last kDepth chunks.
  drain_tail<0>(issuer, smem, tid, v, u, outp);
}

extern "C" void kernel_launch(void* const* d_in, const int* in_sizes, int n_in,
                              void* d_out, int out_size, void* d_ws, size_t ws_size,
                              hipStream_t stream) {
  (void)in_sizes; (void)n_in; (void)out_size; (void)d_ws; (void)ws_size;
  const float* I = (const float*)d_in[0];
  float* spikes  = (float*)d_out;

  dim3 grid(kN / kTileN);   // 256 blocks
  dim3 block(kTileN);       // 128 threads = 4 wave32
  hipLaunchKernelGGL(IzhikevichNeuron_39032662786071_kernel, grid, block,
                     kLdsBytes, stream, I, spikes);
}